// NegativeKLDiv_56367150792858
// MI455X (gfx1250) — compile-verified
//
#include <hip/hip_runtime.h>
#include <hip/hip_bf16.h>

typedef __attribute__((ext_vector_type(2))) float v2f;
typedef __attribute__((ext_vector_type(8))) float v8f;

#define KL_D   4608
#define KL_EPS 1e-9f

// One wave32 handles 4 batches. The 4 independent 4x4xD GEMMs are packed into
// a single 16x16 WMMA tile (diagonal 4x4 blocks are the valid outputs).
// A fragment (16x4 f32): lane L -> row (L&15), K = 2*(L>>4)+{0,1}
// B fragment (4x16 f32): lane L -> col (L&15), K = 2*(L>>4)+{0,1}
// => both fragments load a v2f at  row*D + kk + 2*hi   (perfect partition).
__global__ __launch_bounds__(256) void NegativeKLDiv_wmma_kernel(
    const float* __restrict__ in1, const float* __restrict__ msk1,
    const float* __restrict__ in2, const float* __restrict__ msk2,
    float* __restrict__ out, int bs)
{
    const int lane = threadIdx.x & 31;
    const int wave = blockIdx.x * (blockDim.x >> 5) + (threadIdx.x >> 5);
    const int b0   = wave * 4;                 // first of 4 batches for this wave
    if (b0 >= bs) return;                      // wave-uniform: EXEC stays full

    const int n  = lane & 15;                  // row of p / row of q (4*bb + i)
    const int hi = lane >> 4;                  // half-wave -> K sub-pair

    // row (b0+bb)*4 + (n&3) == 4*b0 + n ; v2f index = row*(D/2) + hi
    const long off2 = (long)(4 * b0 + n) * (KL_D / 2) + hi;
    const v2f* p1 = (const v2f*)in1  + off2;
    const v2f* m1 = (const v2f*)msk1 + off2;
    const v2f* p2 = (const v2f*)in2  + off2;
    const v2f* m2 = (const v2f*)msk2 + off2;

    v8f   c   = {};                            // 16x16 f32 accumulator (8 VGPRs)
    float ent = 0.0f;                          // lane-local entropy partial

    #pragma unroll 4
    for (int kk = 0; kk < KL_D / 4; ++kk) {    // 4 K-values per WMMA step
        v2f a  = __builtin_nontemporal_load(&p1[2 * kk]);
        v2f am = __builtin_nontemporal_load(&m1[2 * kk]);
        v2f b  = __builtin_nontemporal_load(&p2[2 * kk]);
        v2f bm = __builtin_nontemporal_load(&m2[2 * kk]);

        float pa = a[0] * am[0], pb = a[1] * am[1];   // p = input1*mask1
        float qa = b[0] * bm[0], qb = b[1] * bm[1];   // q = input2*mask2

        v2f A;  A[0]  = pa;                       A[1]  = pb;
        v2f Bf; Bf[0] = __logf(qa + KL_EPS);      Bf[1] = __logf(qb + KL_EPS);

        ent = __builtin_fmaf(pa, __logf(pa + KL_EPS), ent);
        ent = __builtin_fmaf(pb, __logf(pb + KL_EPS), ent);

        // D = A(16x4) x B(4x16) + C : accumulate cross[m][n] over K
        c = __builtin_amdgcn_wmma_f32_16x16x4_f32(
                /*neg_a=*/false, A, /*neg_b=*/false, Bf,
                /*c_mod=*/(short)0, c, /*reuse_a=*/false, /*reuse_b=*/false);
    }

    // ent[row n] = lane n partial + lane n+16 partial (full EXEC here)
    float entFull = ent + __shfl_xor(ent, 16, 32);

    // Pre-fetch (full-EXEC) the 4 ent values of this lane's diagonal block:
    // block bb = n>>2 covers rows m = 4*bb .. 4*bb+3, held in lanes m.
    const int bb = n >> 2;
    float e0 = __shfl(entFull, 4 * bb + 0, 32);
    float e1 = __shfl(entFull, 4 * bb + 1, 32);
    float e2 = __shfl(entFull, 4 * bb + 2, 32);
    float e3 = __shfl(entFull, 4 * bb + 3, 32);

    // D layout: VGPR v of lane L holds row m = v + 8*hi, col n = L&15.
    // Diagonal-block rows 4*bb..4*bb+3 live in VGPRs (n&4)+0..3 of lanes with
    // (hi==0 && n<8) or (hi==1 && n>=8).
    const bool active = hi ? (n >= 8) : (n < 8);
    if (active) {
        const int j  = n & 3;                  // column inside 4x4 block
        const int vb = n & 4;                  // VGPR base (0 or 4)
        float c0 = vb ? c[4] : c[0];
        float c1 = vb ? c[5] : c[1];
        float c2 = vb ? c[6] : c[2];
        float c3 = vb ? c[7] : c[3];
        float* o = out + (long)(b0 + bb) * 16 + j;   // out[b][i][j], i stride 4
        o[0]  = c0 - e0;   // -div = cross - ent
        o[4]  = c1 - e1;
        o[8]  = c2 - e2;
        o[12] = c3 - e3;
    }
}

extern "C" void kernel_launch(void* const* d_in, const int* in_sizes, int n_in,
                              void* d_out, int out_size, void* d_ws, size_t ws_size,
                              hipStream_t stream) {
    const float* in1  = (const float*)d_in[0];
    const float* msk1 = (const float*)d_in[1];
    const float* in2  = (const float*)d_in[2];
    const float* msk2 = (const float*)d_in[3];
    float* out = (float*)d_out;

    const int bs    = in_sizes[0] / (4 * KL_D);   // 2048
    const int waves = (bs + 3) / 4;               // 4 batches per wave
    const int wavesPerBlock = 8;                  // 256 threads = 8 wave32
    const int blocks = (waves + wavesPerBlock - 1) / wavesPerBlock;

    NegativeKLDiv_wmma_kernel<<<blocks, 256, 0, stream>>>(
        in1, msk1, in2, msk2, out, bs);
}